// SRLEmbeddings_48172353192026
// MI455X (gfx1250) — compile-verified
//
#include <hip/hip_runtime.h>

// SRL embeddings, MI455X (gfx1250, wave32).
// One workgroup per (b,s) sentence. Build 32x128 weight matrix in LDS
// (row 0 = attention-mask mean, rows 1..24 = per-arg last-valid-token match
// masks scaled by 1/count, rows 25..31 = zero), then compute
// W[32,128] x H[128,768] with V_WMMA_F32_16X16X4_F32 in a single streaming
// pass over hidden_states (memory-bound: ~100 MB @ 23.3 TB/s ~ 4.4 us).

typedef __attribute__((ext_vector_type(2))) float v2f;
typedef __attribute__((ext_vector_type(8))) float v8f;

#define Bb 16
#define Ss 16
#define Ll 128
#define Dd 768
#define Aa 8
#define Tt 5

__global__ __launch_bounds__(256)
void srl_wmma_kernel(const float* __restrict__ hidden,
                     const int*   __restrict__ sent_ids,
                     const int*   __restrict__ masks,
                     const int*   __restrict__ pred_ids,
                     const int*   __restrict__ arg0_ids,
                     const int*   __restrict__ arg1_ids,
                     float*       __restrict__ out)
{
    __shared__ int   sid[Ll];
    __shared__ float W[32][132];   // padded stride -> conflict-free A reads
    __shared__ int   cnt0;

    const int bs   = blockIdx.x;       // 0..B*S-1
    const int tid  = threadIdx.x;      // 0..255
    const int lane = tid & 31;
    const int wv   = tid >> 5;         // 0..7

    if (tid == 0) cnt0 = 0;
    __syncthreads();

    // ---- Phase 1a: sentence ids + attention mask row ----
    if (tid < Ll) {
        int sv = sent_ids[bs * Ll + tid];
        sid[tid] = sv;
        int m = masks[bs * Ll + tid];
        W[0][tid] = (float)m;
        if (m) atomicAdd(&cnt0, 1);
    }
    // zero pad rows 25..31
    for (int i = tid; i < 7 * Ll; i += 256) {
        W[25 + (i >> 7)][i & 127] = 0.0f;
    }
    __syncthreads();

    if (tid < Ll) {
        float c = (float)(cnt0 > 1 ? cnt0 : 1);
        W[0][tid] = W[0][tid] * (1.0f / c);
    }

    // ---- Phase 1b: arg rows 1..24, one wave per row (wave-parallel over L) ----
    for (int r = wv; r < 24; r += 8) {
        const int type = r >> 3;            // 0=pred, 1=arg0, 2=arg1
        const int a    = r & 7;
        const int* ids = (type == 0 ? pred_ids : (type == 1 ? arg0_ids : arg1_ids))
                         + ((size_t)bs * Aa + a) * Tt;
        int id[Tt];
        #pragma unroll
        for (int t = 0; t < Tt; ++t) id[t] = ids[t];

        int c[Tt] = {0, 0, 0, 0, 0};
        #pragma unroll
        for (int k = 0; k < Ll / 32; ++k) {
            int sv = sid[lane + 32 * k];
            #pragma unroll
            for (int t = 0; t < Tt; ++t) c[t] += (sv == id[t]) ? 1 : 0;
        }
        // wave32 butterfly reduction -> every lane holds full counts
        #pragma unroll
        for (int off = 16; off > 0; off >>= 1) {
            #pragma unroll
            for (int t = 0; t < Tt; ++t) c[t] += __shfl_xor(c[t], off, 32);
        }
        // last valid token (nonzero id with at least one match)
        int   lid   = -1;      // never matches sid (ids in [0, VOCAB))
        float scale = 0.0f;
        #pragma unroll
        for (int t = 0; t < Tt; ++t) {
            if (id[t] != 0 && c[t] > 0) { lid = id[t]; scale = 1.0f / (float)c[t]; }
        }
        #pragma unroll
        for (int k = 0; k < Ll / 32; ++k) {
            int l = lane + 32 * k;
            W[1 + r][l] = (sid[l] == lid) ? scale : 0.0f;
        }
    }
    __syncthreads();

    // ---- Phase 2: W[32,128] x H[128,768] via V_WMMA_F32_16X16X4_F32 ----
    const float* H    = hidden + (size_t)bs * (Ll * Dd);
    const int    nt0  = wv * 6;              // 6 N-tiles of 16 per wave (48 total)
    const int    mrow = lane & 15;           // A fragment: M = lane%16
    const int    khalf = (lane >> 4) << 1;   // lanes 0-15 -> K {0,1}; 16-31 -> K {2,3}
    const int    ncol = lane & 15;

    v8f acc[2][6] = {};

    for (int kb = 0; kb < Ll; kb += 4) {
        v2f a0, a1;
        a0.x = W[mrow][kb + khalf];
        a0.y = W[mrow][kb + khalf + 1];
        a1.x = W[16 + mrow][kb + khalf];
        a1.y = W[16 + mrow][kb + khalf + 1];

        const float* Hk = H + (size_t)(kb + khalf) * Dd + ncol;
        #pragma unroll
        for (int j = 0; j < 6; ++j) {
            const int n = (nt0 + j) << 4;
            v2f bf;
            bf.x = Hk[n];
            bf.y = Hk[n + Dd];
            acc[0][j] = __builtin_amdgcn_wmma_f32_16x16x4_f32(
                false, a0, false, bf, (short)0, acc[0][j], false, false);
            acc[1][j] = __builtin_amdgcn_wmma_f32_16x16x4_f32(
                false, a1, false, bf, (short)0, acc[1][j], false, false);
        }
    }

    // ---- Phase 3: scatter C/D fragments to the 4 concatenated outputs ----
    float* outS  = out;                                   // [B,S,D]
    float* outP  = out  + (size_t)Bb * Ss * Dd;           // [B,S,A,D]
    float* outA0 = outP + (size_t)Bb * Ss * Aa * Dd;      // [B,S,A,D]
    float* outA1 = outA0 + (size_t)Bb * Ss * Aa * Dd;     // [B,S,A,D]

    const int rhi = (lane >> 4) << 3;   // VGPR e holds row e (lanes 0-15) / e+8
    #pragma unroll
    for (int mt = 0; mt < 2; ++mt) {
        #pragma unroll
        for (int j = 0; j < 6; ++j) {
            const int n = (nt0 + j) * 16 + ncol;
            #pragma unroll
            for (int e = 0; e < 8; ++e) {
                const int row = mt * 16 + rhi + e;   // 0..31
                const float v = acc[mt][j][e];
                if (row == 0) {
                    outS[(size_t)bs * Dd + n] = v;
                } else if (row <= 24) {
                    const int r = row - 1;           // 0..23
                    float* dst = ((r >> 3) == 0) ? outP
                               : (((r >> 3) == 1) ? outA0 : outA1);
                    dst[((size_t)bs * Aa + (r & 7)) * Dd + n] = v;
                }
            }
        }
    }
}

extern "C" void kernel_launch(void* const* d_in, const int* in_sizes, int n_in,
                              void* d_out, int out_size, void* d_ws, size_t ws_size,
                              hipStream_t stream) {
    const float* hidden = (const float*)d_in[0];
    const int*   sids   = (const int*)d_in[1];
    const int*   msk    = (const int*)d_in[2];
    const int*   pred   = (const int*)d_in[3];
    const int*   a0     = (const int*)d_in[4];
    const int*   a1     = (const int*)d_in[5];
    float*       out    = (float*)d_out;

    dim3 grid(Bb * Ss);
    dim3 block(256);
    hipLaunchKernelGGL(srl_wmma_kernel, grid, block, 0, stream,
                       hidden, sids, msk, pred, a0, a1, out);
}